// GraphEncoder_1288490189003
// MI455X (gfx1250) — compile-verified
//
#include <hip/hip_runtime.h>
#include <math.h>

typedef float v2f __attribute__((ext_vector_type(2)));
typedef float v8f __attribute__((ext_vector_type(8)));

constexpr int NN   = 100000;   // nodes
constexpr int EE   = 800000;   // edges (without self loops)
constexpr int ETOT = EE + NN;  // edges + self loops
constexpr int DD   = 64;       // per-head dim
constexpr int FF   = 128;      // H*D
constexpr int GG   = 1000;     // graphs

// ---------------- utility fills ----------------
__global__ void fill_f32(float* __restrict__ p, float v, long long n) {
  long long i = blockIdx.x * (long long)blockDim.x + threadIdx.x;
  if (i < n) p[i] = v;
}
__global__ void fill_u32(unsigned* __restrict__ p, unsigned v, long long n) {
  long long i = blockIdx.x * (long long)blockDim.x + threadIdx.x;
  if (i < n) p[i] = v;
}

// ---------------- embedding gather ----------------
__global__ void gather_embed(const int* __restrict__ x, const float* __restrict__ emb,
                             float* __restrict__ feat) {
  long long i = blockIdx.x * (long long)blockDim.x + threadIdx.x;
  if (i >= (long long)NN * DD) return;
  int n = (int)(i >> 6);
  int d = (int)(i & 63);
  feat[i] = emb[(long long)x[n] * DD + d];
}

// ---------------- fp32 WMMA GEMM: Out[M,Nc] = A[M,K] @ W[K,Nc] (+bias) ----------------
// One wave computes a 16x16 tile via v_wmma_f32_16x16x4_f32, looping K in steps of 4.
// blockDim = (32,4): 4 waves handle 4 adjacent N-tiles.  M,Nc multiples of 16 (Nc mult of 64).
__global__ void wmma_gemm_f32(const float* __restrict__ A, const float* __restrict__ W,
                              const float* __restrict__ bias, float* __restrict__ Out,
                              int M, int K, int Nc) {
  const int lane  = threadIdx.x;       // 0..31
  const int row16 = lane & 15;
  const int ksel  = lane >> 4;         // 0 or 1
  const int n0 = (blockIdx.x * 4 + threadIdx.y) * 16;
  const int m0 = blockIdx.y * 16;
  if (n0 >= Nc || m0 >= M) return;     // wave-uniform

  const float* __restrict__ arow = A + (long long)(m0 + row16) * K;
  v8f acc = {};
  for (int k0 = 0; k0 < K; k0 += 4) {
    const int ka = k0 + ksel * 2;
    // A 16x4 fragment: lanes 0-15 -> K=k0..k0+1, lanes 16-31 -> K=k0+2..k0+3
    v2f a;
    a.x = arow[ka];
    a.y = arow[ka + 1];
    // B 4x16 fragment: same K split, column = n0+row16
    v2f b;
    b.x = W[(long long)ka       * Nc + n0 + row16];
    b.y = W[(long long)(ka + 1) * Nc + n0 + row16];
    acc = __builtin_amdgcn_wmma_f32_16x16x4_f32(
        /*neg_a=*/false, a, /*neg_b=*/false, b,
        /*c_mod=*/(short)0, acc, /*reuse_a=*/false, /*reuse_b=*/false);
  }
  // C/D layout: VGPR r, lanes 0-15 -> M=r, lanes 16-31 -> M=r+8; N = lane&15
#pragma unroll
  for (int r = 0; r < 8; ++r) {
    const int m = m0 + r + ksel * 8;
    const int n = n0 + row16;
    float v = acc[r];
    if (bias) v += bias[n];
    Out[(long long)m * Nc + n] = v;
  }
}

// ---------------- attention logits: als/ald[n,h] = <h[n,h,:], a_src/dst[h,:]> ----------------
__global__ void attn_logits(const float* __restrict__ h, const float* __restrict__ a_src,
                            const float* __restrict__ a_dst, float* __restrict__ als,
                            float* __restrict__ ald) {
  long long i = blockIdx.x * (long long)blockDim.x + threadIdx.x;  // over N*2
  if (i >= (long long)NN * 2) return;
  const int n  = (int)(i >> 1);
  const int hh = (int)(i & 1);
  const float* __restrict__ hv = h + (long long)n * FF + hh * DD;
  const float* __restrict__ as = a_src + hh * DD;
  const float* __restrict__ ad = a_dst + hh * DD;
  float s = 0.f, d = 0.f;
#pragma unroll 8
  for (int k = 0; k < DD; ++k) {
    const float v = hv[k];
    s += v * as[k];
    d += v * ad[k];
  }
  als[i] = s;
  ald[i] = d;
}

// ---------------- monotone float<->uint key for atomic max ----------------
__device__ __forceinline__ unsigned fkey(float f) {
  unsigned u = __float_as_uint(f);
  return (u & 0x80000000u) ? ~u : (u | 0x80000000u);
}
__device__ __forceinline__ float funkey(unsigned k) {
  unsigned u = (k & 0x80000000u) ? (k & 0x7FFFFFFFu) : ~k;
  return __uint_as_float(u);
}

// ---------------- edge pass 1: logits + segment max ----------------
__global__ void edge_logit_max(const int* __restrict__ src, const int* __restrict__ dst,
                               const float* __restrict__ als, const float* __restrict__ ald,
                               float* __restrict__ ebuf, unsigned* __restrict__ mkey) {
  long long e = blockIdx.x * (long long)blockDim.x + threadIdx.x;
  if (e >= ETOT) return;
  const int s = (e < EE) ? src[e] : (int)(e - EE);
  const int d = (e < EE) ? dst[e] : (int)(e - EE);
#pragma unroll
  for (int hh = 0; hh < 2; ++hh) {
    float v = als[(long long)s * 2 + hh] + ald[(long long)d * 2 + hh];
    v = (v > 0.f) ? v : 0.2f * v;  // leaky_relu, slope 0.2
    ebuf[e * 2 + hh] = v;
    atomicMax(&mkey[(long long)d * 2 + hh], fkey(v));
  }
}

// ---------------- edge pass 2: w = exp(e - m[dst]); denom += w ----------------
__global__ void edge_exp_den(const int* __restrict__ dst, float* __restrict__ ebuf,
                             const unsigned* __restrict__ mkey, float* __restrict__ den) {
  long long e = blockIdx.x * (long long)blockDim.x + threadIdx.x;
  if (e >= ETOT) return;
  const int d = (e < EE) ? dst[e] : (int)(e - EE);
#pragma unroll
  for (int hh = 0; hh < 2; ++hh) {
    const float m = funkey(mkey[(long long)d * 2 + hh]);
    const float w = expf(ebuf[e * 2 + hh] - m);
    ebuf[e * 2 + hh] = w;
    atomicAdd(&den[(long long)d * 2 + hh], w);
  }
}

// ---------------- edge pass 3: normalize w ----------------
__global__ void edge_norm(const int* __restrict__ dst, float* __restrict__ ebuf,
                          const float* __restrict__ den) {
  long long e = blockIdx.x * (long long)blockDim.x + threadIdx.x;
  if (e >= ETOT) return;
  const int d = (e < EE) ? dst[e] : (int)(e - EE);
#pragma unroll
  for (int hh = 0; hh < 2; ++hh)
    ebuf[e * 2 + hh] /= den[(long long)d * 2 + hh];
}

// ---------------- edge pass 4: out[dst,f] += h[src,f] * w[e,h(f)] ----------------
// 128 consecutive threads per edge => coalesced gather + coalesced atomics.
__global__ void edge_aggregate(const int* __restrict__ src, const int* __restrict__ dst,
                               const float* __restrict__ ebuf, const float* __restrict__ hin,
                               float* __restrict__ out) {
  long long tid = blockIdx.x * (long long)blockDim.x + threadIdx.x;
  if (tid >= (long long)ETOT * FF) return;
  const long long e = tid >> 7;
  const int f = (int)(tid & 127);
  const int s = (e < EE) ? src[e] : (int)(e - EE);
  const int d = (e < EE) ? dst[e] : (int)(e - EE);
  const int hh = f >> 6;
  const float w = ebuf[e * 2 + hh];
  atomicAdd(&out[(long long)d * FF + f], hin[(long long)s * FF + f] * w);
}

// ---------------- bias + leaky_relu (slope 0.01) over N x 128 ----------------
__global__ void bias_act(float* __restrict__ out, const float* __restrict__ b, float slope) {
  long long i = blockIdx.x * (long long)blockDim.x + threadIdx.x;
  if (i >= (long long)NN * FF) return;
  const int f = (int)(i & 127);
  float v = out[i] + b[f];
  out[i] = (v > 0.f) ? v : slope * v;
}

// ---------------- per-graph mean pooling ----------------
__global__ void pool_sum(const float* __restrict__ h, const int* __restrict__ batch,
                         float* __restrict__ sums, float* __restrict__ counts) {
  long long i = blockIdx.x * (long long)blockDim.x + threadIdx.x;
  if (i >= (long long)NN * DD) return;
  const int n = (int)(i >> 6);
  const int d = (int)(i & 63);
  const int g = batch[n];
  atomicAdd(&sums[(long long)g * DD + d], h[i]);
  if (d == 0) atomicAdd(&counts[g], 1.0f);
}
__global__ void pool_div(const float* __restrict__ sums, const float* __restrict__ counts,
                         float* __restrict__ z) {
  int i = blockIdx.x * blockDim.x + threadIdx.x;
  if (i >= GG * DD) return;
  const int g = i >> 6;
  float c = counts[g];
  c = (c > 1.f) ? c : 1.f;
  z[i] = sums[i] / c;
}

// ---------------- host orchestration ----------------
extern "C" void kernel_launch(void* const* d_in, const int* in_sizes, int n_in,
                              void* d_out, int out_size, void* d_ws, size_t ws_size,
                              hipStream_t stream) {
  (void)in_sizes; (void)n_in; (void)out_size; (void)ws_size;
  const int*   x     = (const int*)d_in[0];
  const int*   ei    = (const int*)d_in[1];
  const int*   batch = (const int*)d_in[2];
  const float* emb   = (const float*)d_in[3];
  const float* W1    = (const float*)d_in[4];
  const float* as1   = (const float*)d_in[5];
  const float* ad1   = (const float*)d_in[6];
  const float* b1    = (const float*)d_in[7];
  const float* W2    = (const float*)d_in[8];
  const float* as2   = (const float*)d_in[9];
  const float* ad2   = (const float*)d_in[10];
  const float* b2    = (const float*)d_in[11];
  const float* Wp    = (const float*)d_in[12];
  const float* bp    = (const float*)d_in[13];

  const int* src = ei;        // edge_index[0]
  const int* dst = ei + EE;   // edge_index[1]

  float* hOut = (float*)d_out;                 // N x 64
  float* z    = hOut + (long long)NN * DD;     // G x 64

  // workspace carve-up (floats)
  float*    ws    = (float*)d_ws;
  float*    bufA  = ws;                                  // N x 128 (feat / agg2)
  float*    bufB  = bufA + (long long)NN * FF;           // N x 128 (h = x@W)
  float*    bufC  = bufB + (long long)NN * FF;           // N x 128 (agg1 / layer1 out)
  float*    als   = bufC + (long long)NN * FF;           // N x 2
  float*    ald   = als + (long long)NN * 2;             // N x 2
  unsigned* mkey  = (unsigned*)(ald + (long long)NN * 2);// N x 2
  float*    den   = (float*)mkey + (long long)NN * 2;    // N x 2
  float*    ebuf  = den + (long long)NN * 2;             // ETOT x 2
  float*    sums  = ebuf + (long long)ETOT * 2;          // G x 64
  float*    cnts  = sums + (long long)GG * DD;           // G

  auto blk = [](long long n, int t) { return (unsigned)((n + t - 1) / t); };

  // --- embedding gather ---
  gather_embed<<<blk((long long)NN * DD, 256), 256, 0, stream>>>(x, emb, bufA);

  // ======================= GAT layer 1 =======================
  wmma_gemm_f32<<<dim3(FF / 64, NN / 16), dim3(32, 4), 0, stream>>>(bufA, W1, nullptr, bufB, NN, DD, FF);
  attn_logits<<<blk((long long)NN * 2, 256), 256, 0, stream>>>(bufB, as1, ad1, als, ald);
  fill_u32<<<blk((long long)NN * 2, 256), 256, 0, stream>>>(mkey, 0u, (long long)NN * 2);
  edge_logit_max<<<blk(ETOT, 256), 256, 0, stream>>>(src, dst, als, ald, ebuf, mkey);
  fill_f32<<<blk((long long)NN * 2, 256), 256, 0, stream>>>(den, 0.f, (long long)NN * 2);
  edge_exp_den<<<blk(ETOT, 256), 256, 0, stream>>>(dst, ebuf, mkey, den);
  edge_norm<<<blk(ETOT, 256), 256, 0, stream>>>(dst, ebuf, den);
  fill_f32<<<blk((long long)NN * FF, 256), 256, 0, stream>>>(bufC, 0.f, (long long)NN * FF);
  edge_aggregate<<<blk((long long)ETOT * FF, 256), 256, 0, stream>>>(src, dst, ebuf, bufB, bufC);
  bias_act<<<blk((long long)NN * FF, 256), 256, 0, stream>>>(bufC, b1, 0.01f);

  // ======================= GAT layer 2 =======================
  wmma_gemm_f32<<<dim3(FF / 64, NN / 16), dim3(32, 4), 0, stream>>>(bufC, W2, nullptr, bufB, NN, FF, FF);
  attn_logits<<<blk((long long)NN * 2, 256), 256, 0, stream>>>(bufB, as2, ad2, als, ald);
  fill_u32<<<blk((long long)NN * 2, 256), 256, 0, stream>>>(mkey, 0u, (long long)NN * 2);
  edge_logit_max<<<blk(ETOT, 256), 256, 0, stream>>>(src, dst, als, ald, ebuf, mkey);
  fill_f32<<<blk((long long)NN * 2, 256), 256, 0, stream>>>(den, 0.f, (long long)NN * 2);
  edge_exp_den<<<blk(ETOT, 256), 256, 0, stream>>>(dst, ebuf, mkey, den);
  edge_norm<<<blk(ETOT, 256), 256, 0, stream>>>(dst, ebuf, den);
  fill_f32<<<blk((long long)NN * FF, 256), 256, 0, stream>>>(bufA, 0.f, (long long)NN * FF);
  edge_aggregate<<<blk((long long)ETOT * FF, 256), 256, 0, stream>>>(src, dst, ebuf, bufB, bufA);
  bias_act<<<blk((long long)NN * FF, 256), 256, 0, stream>>>(bufA, b2, 0.01f);

  // ======================= projection (fused bias) =======================
  wmma_gemm_f32<<<dim3(DD / 64, NN / 16), dim3(32, 4), 0, stream>>>(bufA, Wp, bp, hOut, NN, FF, DD);

  // ======================= mean pool per graph =======================
  fill_f32<<<blk((long long)GG * DD + GG, 256), 256, 0, stream>>>(sums, 0.f, (long long)GG * DD + GG);
  pool_sum<<<blk((long long)NN * DD, 256), 256, 0, stream>>>(hOut, batch, sums, cnts);
  pool_div<<<blk((long long)GG * DD, 256), 256, 0, stream>>>(sums, cnts, z);
}